// RoIAlign_18691697672635
// MI455X (gfx1250) — compile-verified
//
#include <hip/hip_runtime.h>

// ---------------------------------------------------------------------------
// 3D ROI Align, MI455X (gfx1250).
// feat (N=2, C=32, D=64, H=128, W=128) f32; bboxes (K,7) f32;
// out (K, C, 16, 16, 16) f32.
// Block = one (k, zo) slab; 256 threads = (yo,xo) plane; loop over C.
// Separable trilinear weights precomputed per-axis into LDS.
// CDNA5 paths: TDM tensor_load_to_lds bbox staging (+ s_wait_tensorcnt),
// global_prefetch_b8, non-temporal output stores.
// ---------------------------------------------------------------------------

constexpr int Cc = 32, Dd = 64, Hh = 128, Ww = 128;
constexpr int HW  = Hh * Ww;        // 16384
constexpr int DHW = Dd * HW;        // 1048576
constexpr int Do_ = 16, Ho_ = 16, Wo_ = 16;
constexpr int OSLAB = Do_ * Ho_ * Wo_;  // 4096 elems per (k,c)

typedef unsigned int u32x4 __attribute__((ext_vector_type(4)));
typedef int          i32x8 __attribute__((ext_vector_type(8)));
typedef int          i32x4 __attribute__((ext_vector_type(4)));

static __device__ __forceinline__ int iclamp(int v, int lo, int hi) {
    return v < lo ? lo : (v > hi ? hi : v);
}

#if defined(__has_builtin)
#if __has_builtin(__builtin_amdgcn_tensor_load_to_lds)
#define RA_HAVE_TDM 1
#endif
#endif

#ifdef RA_HAVE_TDM
// Minimal 1-D D# (count=1, data_size=4B, tile_dim0=ndw dwords), packed per
// CDNA5 ISA. TDM ignores EXEC; single-lane issue.
static __device__ __forceinline__ void tdm_stage_dwords(const void* gsrc,
                                                        unsigned lds_byte_off,
                                                        int ndw) {
    unsigned long long ga = (unsigned long long)(uintptr_t)gsrc;
    u32x4 g0;
    g0.x = 1u;                                        // count=1 (valid), user mode
    g0.y = lds_byte_off;                              // lds_addr (bytes)
    g0.z = (unsigned)(ga & 0xFFFFFFFFull);            // global_addr[31:0]
    g0.w = (unsigned)((ga >> 32) & 0x1FFFFFFull)      // global_addr[56:32]
           | (2u << 30);                              // type = 2 ("image")
    i32x8 g1;
    g1[0] = (2 << 16);                                // data_size = 4B
    g1[1] = (ndw & 0xFFFF) << 16;                     // tensor_dim0[15:0]
    g1[2] = (1 << 16);                                // dim0 hi=0; tensor_dim1=1
    g1[3] = (ndw & 0xFFFF) << 16;                     // tile_dim0 = ndw
    g1[4] = 0;                                        // tile_dim1=0, tile_dim2=0
    g1[5] = ndw;                                      // tensor_dim0_stride lo32
    g1[6] = 0;
    g1[7] = 0;
    i32x4 gz = {0, 0, 0, 0};
#if __clang_major__ >= 23
    i32x8 gz8 = {0, 0, 0, 0, 0, 0, 0, 0};
    __builtin_amdgcn_tensor_load_to_lds(g0, g1, gz, gz, gz8, 0);
#else
    __builtin_amdgcn_tensor_load_to_lds(g0, g1, gz, gz, 0);
#endif
#if __has_builtin(__builtin_amdgcn_s_wait_tensorcnt)
    __builtin_amdgcn_s_wait_tensorcnt(0);
#else
    asm volatile("s_wait_tensorcnt 0x0" ::: "memory");
#endif
}
#endif // RA_HAVE_TDM

__global__ __launch_bounds__(256) void roi_align3d_kernel(
    const float* __restrict__ feat, const float* __restrict__ bboxes,
    float* __restrict__ out, int K) {
    __shared__ float ldsB[8];                 // bbox row: b, p1xyz, p2xyz
    __shared__ int   sx0[16], sx1[16];
    __shared__ float swx0[16], swx1[16];
    __shared__ int   sy0[16], sy1[16];        // pre-multiplied by W
    __shared__ float swy0[16], swy1[16];
    __shared__ int   szoff0, szoff1, sbatch;  // pre-multiplied by HW
    __shared__ float swz0, swz1;

    const int tid = threadIdx.x;
    const int k   = blockIdx.x >> 4;
    const int zo  = blockIdx.x & 15;

    // ---- Stage this ROI's bbox row (7 dwords) into LDS -----------------
#ifdef RA_HAVE_TDM
    if (tid == 0) {
        unsigned lds_off = (unsigned)(unsigned long long)(
            __attribute__((address_space(3))) void*)&ldsB[0];
        tdm_stage_dwords(bboxes + (size_t)k * 7, lds_off, 7);
    }
#else
    if (tid < 7) ldsB[tid] = bboxes[(size_t)k * 7 + tid];
#endif
    __syncthreads();

    // ---- Per-axis separable tables -------------------------------------
    if (tid < 16) {                         // x axis
        const float p1 = ldsB[1], p2 = ldsB[4];
        const float s = (p2 - p1) * (1.0f / (Ww - 1));
        const float t = (p2 + p1 + 1.0f) * (1.0f / Ww) - 1.0f;
        const float bse = (2.0f * tid + 1.0f) * (1.0f / Wo_) - 1.0f;
        const float ic = ((s * bse + t + 1.0f) * Ww - 1.0f) * 0.5f;
        const float f0 = floorf(ic);
        const float fr = ic - f0;
        const int i0 = (int)f0, i1 = i0 + 1;
        sx0[tid]  = iclamp(i0, 0, Ww - 1);
        sx1[tid]  = iclamp(i1, 0, Ww - 1);
        swx0[tid] = (i0 >= 0 && i0 < Ww) ? (1.0f - fr) : 0.0f;
        swx1[tid] = (i1 >= 0 && i1 < Ww) ? fr : 0.0f;
    } else if (tid < 32) {                  // y axis
        const int j = tid - 16;
        const float p1 = ldsB[2], p2 = ldsB[5];
        const float s = (p2 - p1) * (1.0f / (Hh - 1));
        const float t = (p2 + p1 + 1.0f) * (1.0f / Hh) - 1.0f;
        const float bse = (2.0f * j + 1.0f) * (1.0f / Ho_) - 1.0f;
        const float ic = ((s * bse + t + 1.0f) * Hh - 1.0f) * 0.5f;
        const float f0 = floorf(ic);
        const float fr = ic - f0;
        const int i0 = (int)f0, i1 = i0 + 1;
        sy0[j]  = iclamp(i0, 0, Hh - 1) * Ww;
        sy1[j]  = iclamp(i1, 0, Hh - 1) * Ww;
        swy0[j] = (i0 >= 0 && i0 < Hh) ? (1.0f - fr) : 0.0f;
        swy1[j] = (i1 >= 0 && i1 < Hh) ? fr : 0.0f;
    } else if (tid == 32) {                 // z axis (block-uniform zo)
        const float p1 = ldsB[3], p2 = ldsB[6];
        const float s = (p2 - p1) * (1.0f / (Dd - 1));
        const float t = (p2 + p1 + 1.0f) * (1.0f / Dd) - 1.0f;
        const float bse = (2.0f * zo + 1.0f) * (1.0f / Do_) - 1.0f;
        const float ic = ((s * bse + t + 1.0f) * Dd - 1.0f) * 0.5f;
        const float f0 = floorf(ic);
        const float fr = ic - f0;
        const int i0 = (int)f0, i1 = i0 + 1;
        szoff0 = iclamp(i0, 0, Dd - 1) * HW;
        szoff1 = iclamp(i1, 0, Dd - 1) * HW;
        swz0 = (i0 >= 0 && i0 < Dd) ? (1.0f - fr) : 0.0f;
        swz1 = (i1 >= 0 && i1 < Dd) ? fr : 0.0f;
        sbatch = (int)ldsB[0];
    }
    __syncthreads();

    // ---- Main loop over channels ---------------------------------------
    const int xo = tid & 15;
    const int yo = tid >> 4;
    const int   x0 = sx0[xo], x1 = sx1[xo];
    const float wx0 = swx0[xo], wx1 = swx1[xo];
    const int   y0 = sy0[yo], y1 = sy1[yo];
    const float wy0 = swy0[yo], wy1 = swy1[yo];
    const int   z0 = szoff0, z1 = szoff1;
    const float wz0 = swz0, wz1 = swz1;

    const int r00 = z0 + y0, r01 = z0 + y1;
    const int r10 = z1 + y0, r11 = z1 + y1;

    const float* __restrict__ fb = feat + (size_t)sbatch * ((size_t)Cc * DHW);
    float* __restrict__ op =
        out + ((size_t)k * Cc * Do_ + zo) * (Ho_ * Wo_) + tid;

    for (int c = 0; c < Cc; ++c) {
        const float* __restrict__ fc = fb + (size_t)c * DHW;
        const float v000 = fc[r00 + x0], v001 = fc[r00 + x1];
        const float v010 = fc[r01 + x0], v011 = fc[r01 + x1];
        const float v100 = fc[r10 + x0], v101 = fc[r10 + x1];
        const float v110 = fc[r11 + x0], v111 = fc[r11 + x1];
        if (c + 1 < Cc) __builtin_prefetch(fc + DHW + r00 + x0, 0, 0);
        const float a0 = wx0 * v000 + wx1 * v001;
        const float a1 = wx0 * v010 + wx1 * v011;
        const float a2 = wx0 * v100 + wx1 * v101;
        const float a3 = wx0 * v110 + wx1 * v111;
        const float b0 = wy0 * a0 + wy1 * a1;
        const float b1 = wy0 * a2 + wy1 * a3;
        const float r  = wz0 * b0 + wz1 * b1;
        __builtin_nontemporal_store(r, op);   // write-once output: NT, keep L2 for feat
        op += OSLAB;                          // advance one channel slab (4096)
    }
}

extern "C" void kernel_launch(void* const* d_in, const int* in_sizes, int n_in,
                              void* d_out, int out_size, void* d_ws,
                              size_t ws_size, hipStream_t stream) {
    (void)n_in; (void)out_size; (void)d_ws; (void)ws_size;
    const float* feat   = (const float*)d_in[0];
    const float* bboxes = (const float*)d_in[1];
    float* out = (float*)d_out;
    const int K = in_sizes[1] / 7;           // bboxes is (K, 7)
    dim3 grid((unsigned)(K * Do_));          // one block per (k, zo)
    roi_align3d_kernel<<<grid, 256, 0, stream>>>(feat, bboxes, out, K);
}